// TransformedConv2D_46454366273942
// MI455X (gfx1250) — compile-verified
//
#include <hip/hip_runtime.h>

// ---------------------------------------------------------------------------
// TransformedConv2D for gfx1250 (MI455X) using fp32 WMMA (V_WMMA_F32_16X16X4_F32).
// Implicit GEMM, K reordered as tap*Cin + c so the channel loop is affine:
// fully unrolled ds_load_b32 with immediate offsets, zero per-iteration decode.
// ---------------------------------------------------------------------------

typedef float v2f __attribute__((ext_vector_type(2)));
typedef float v8f __attribute__((ext_vector_type(8)));

#define HH    128
#define WW    128
#define CIN   64
#define COUT  64
#define TAPS  9
#define KSZ   (CIN * TAPS)        // 576

// Part-1 workgroup tile geometry
#define SPW    64                 // spatial columns per workgroup
#define HALO1  (SPW + 2)          // 66
#define CROW   (3 * HALO1)        // 198 floats per channel (3 halo rows)

static __device__ __forceinline__ v8f wmma4(v2f a, v2f b, v8f c) {
  return __builtin_amdgcn_wmma_f32_16x16x4_f32(false, a, false, b, (short)0, c,
                                               false, false);
}

// ---------------------------------------------------------------------------
// Repack weight OIHW[Cout][Cin][3][3] -> wp[tap][Cout][Cin] so A fragments are
// contiguous float2 in the K(=channel) dimension.
// ---------------------------------------------------------------------------
__global__ __launch_bounds__(256)
void pack_weight(const float* __restrict__ wgt, float* __restrict__ wp) {
  const int i = blockIdx.x * 256 + threadIdx.x;   // over tap*COUT*CIN
  if (i < TAPS * COUT * CIN) {
    const int c   = i & (CIN - 1);
    const int co  = (i >> 6) & (COUT - 1);
    const int tap = i >> 12;
    wp[i] = wgt[(co * CIN + c) * TAPS + tap];
  }
}

// ---------------------------------------------------------------------------
// Part 1: full conv over samples [0, i_first).
// grid = (WW/SPW, HH, i_first), block = 128 (4 waves).
// Wave (cg,sg) computes a 32x32 tile: 2 Cout tiles x 2 spatial tiles,
// 4 f32 accumulators, K loop = 9 taps x 16 channel-quads.
// ---------------------------------------------------------------------------
template <bool PACKED>
__global__ __launch_bounds__(128)
void conv_full_wmma(const float* __restrict__ x, const float* __restrict__ wsrc,
                    const float* __restrict__ bias, float* __restrict__ out)
{
  __shared__ float lds[CIN * CROW];   // [c][kh(3)][66] = 50688 B

  const int w0  = blockIdx.x * SPW;
  const int h   = blockIdx.y;
  const int n   = blockIdx.z;
  const int tid = threadIdx.x;

  // ---- stage input halo tile: rows h-1..h+1, cols w0-1..w0+SPW, all Cin ----
  const float* xn = x + (size_t)n * CIN * HH * WW;
  for (int idx = tid; idx < CIN * CROW; idx += 128) {
    int c   = idx / CROW;
    int rem = idx - c * CROW;
    int r   = rem / HALO1;
    int col = rem - r * HALO1;
    int hh  = h - 1 + r;
    int ww  = w0 - 1 + col;
    float v = 0.0f;
    if ((unsigned)hh < HH && (unsigned)ww < WW)
      v = xn[((size_t)c * HH + hh) * WW + ww];
    lds[idx] = v;
  }
  __syncthreads();

  const int wave   = tid >> 5;
  const int lane   = tid & 31;
  const int laneLo = lane & 15;
  const int laneHi = lane >> 4;        // 0/1
  const int cg     = wave >> 1;        // co group: 0 -> co 0..31, 1 -> 32..63
  const int sg     = wave & 1;         // sp group: 0 -> sp 0..31, 1 -> 32..63
  const int co0    = cg * 32;
  const int sp0    = sg * 32;

  v8f acc[2][2];
#pragma unroll
  for (int i = 0; i < 2; ++i)
#pragma unroll
    for (int j = 0; j < 2; ++j) acc[i][j] = (v8f){0,0,0,0,0,0,0,0};

  // A row base (per-lane M row); K element = channel c.
  const float* a0base;
  const float* a1base;
  if (PACKED) {
    a0base = wsrc + (size_t)(co0 + laneLo) * CIN + 2 * laneHi;      // [tap][co][c]
    a1base = a0base + 16 * CIN;
    __builtin_prefetch(a0base, 0, 1);                               // global_prefetch_b8
  } else {
    a0base = wsrc + ((size_t)(co0 + laneLo) * CIN + 2 * laneHi) * TAPS;  // OIHW
    a1base = a0base + (size_t)16 * CIN * TAPS;
  }

#pragma unroll
  for (int kh = 0; kh < 3; ++kh) {
#pragma unroll
    for (int kw = 0; kw < 3; ++kw) {
      const int tap = kh * 3 + kw;
      const float* a0p = PACKED ? (a0base + (size_t)tap * COUT * CIN) : (a0base + tap);
      const float* a1p = PACKED ? (a1base + (size_t)tap * COUT * CIN) : (a1base + tap);
      // B base: element (c = 2*laneHi, kh, col) ; per-channel stride CROW.
      const float* b0p = lds + (2 * laneHi) * CROW + kh * HALO1 + sp0 + laneLo + kw;
      const float* b1p = b0p + 16;

#pragma unroll
      for (int cb = 0; cb < CIN; cb += 4) {   // 16 fully-unrolled K-quads
        v2f a0, a1, b0, b1;
        if (PACKED) {
          a0 = *reinterpret_cast<const v2f*>(a0p + cb);
          a1 = *reinterpret_cast<const v2f*>(a1p + cb);
        } else {
          a0.x = a0p[cb * TAPS]; a0.y = a0p[cb * TAPS + TAPS];
          a1.x = a1p[cb * TAPS]; a1.y = a1p[cb * TAPS + TAPS];
        }
        b0.x = b0p[cb * CROW]; b0.y = b0p[cb * CROW + CROW];
        b1.x = b1p[cb * CROW]; b1.y = b1p[cb * CROW + CROW];

        acc[0][0] = wmma4(a0, b0, acc[0][0]);
        acc[0][1] = wmma4(a0, b1, acc[0][1]);
        acc[1][0] = wmma4(a1, b0, acc[1][0]);
        acc[1][1] = wmma4(a1, b1, acc[1][1]);
      }
    }
  }

  // ---- store: D 16x16 f32 layout: VGPR r -> row r+8*laneHi, col laneLo ----
  const size_t plane = (size_t)HH * WW;
  float* op = out + (size_t)n * COUT * plane + (size_t)h * WW + (w0 + laneLo);
#pragma unroll
  for (int ci = 0; ci < 2; ++ci) {
#pragma unroll
    for (int si = 0; si < 2; ++si) {
      const int cobase = co0 + ci * 16 + 8 * laneHi;
      const int col    = sp0 + si * 16;
#pragma unroll
      for (int r = 0; r < 8; ++r) {
        const int co = cobase + r;
        float v = acc[ci][si][r];
        if (n == 0) v += bias[co];          // bias only on output sample 0
        op[(size_t)co * plane + col] = v;
      }
    }
  }
}

// ---------------------------------------------------------------------------
// Part 2: per-term single-channel conv (store-bandwidth bound).
// K=9 padded to 12 -> 3 WMMA steps. grid = (WW/32, HH, T), block = 256.
// ---------------------------------------------------------------------------
__global__ __launch_bounds__(256)
void conv_terms_wmma(const float* __restrict__ x, const float* __restrict__ wgt,
                     const float* __restrict__ bias,
                     const int* __restrict__ term_err,
                     const int* __restrict__ term_feat,
                     float* __restrict__ out, int i_first)
{
  __shared__ float lds_in[3 * 34];

  const int w0  = blockIdx.x * 32;
  const int h   = blockIdx.y;
  const int t   = blockIdx.z;
  const int tid = threadIdx.x;

  const int n = term_err[t];
  const int f = term_feat[t];

  const float* xp = x + ((size_t)n * CIN + f) * (HH * WW);
  for (int idx = tid; idx < 3 * 34; idx += 256) {
    int r   = idx / 34;
    int col = idx - r * 34;
    int hh  = h - 1 + r;
    int ww  = w0 - 1 + col;
    float v = 0.0f;
    if ((unsigned)hh < HH && (unsigned)ww < WW) v = xp[(size_t)hh * WW + ww];
    lds_in[idx] = v;
  }
  __syncthreads();

  const int wave    = tid >> 5;
  const int lane    = tid & 31;
  const int laneLo  = lane & 15;
  const int laneHi  = lane >> 4;
  const int co_base = (wave & 3) * 16;
  const int sp_off  = (wave >> 2) * 16;

  const float* wrow = wgt + (size_t)(co_base + laneLo) * KSZ + f * TAPS;

  v8f acc = {0, 0, 0, 0, 0, 0, 0, 0};

#pragma unroll
  for (int kb = 0; kb < 12; kb += 4) {
    const int k0 = kb + 2 * laneHi;
    v2f a, b;
    a.x = (k0 < 9) ? wrow[k0] : 0.0f;
    a.y = (k0 + 1 < 9) ? wrow[k0 + 1] : 0.0f;
    b.x = 0.0f;
    b.y = 0.0f;
    if (k0 < 9) {
      int kh = k0 / 3, kw = k0 - kh * 3;
      b.x = lds_in[kh * 34 + sp_off + laneLo + kw];
    }
    if (k0 + 1 < 9) {
      int k1 = k0 + 1;
      int kh = k1 / 3, kw = k1 - kh * 3;
      b.y = lds_in[kh * 34 + sp_off + laneLo + kw];
    }
    acc = wmma4(a, b, acc);
  }

  const int osample = i_first + t;
  float* op = out + (size_t)osample * COUT * (HH * WW) + (size_t)h * WW
                  + (w0 + sp_off + laneLo);
#pragma unroll
  for (int r = 0; r < 8; ++r) {
    const int co = co_base + r + 8 * laneHi;
    float v = acc[r];
    if (osample == 0) v += bias[co];   // only if i_first == 0
    op[(size_t)co * (HH * WW)] = v;
  }
}

// ---------------------------------------------------------------------------
extern "C" void kernel_launch(void* const* d_in, const int* in_sizes, int n_in,
                              void* d_out, int out_size, void* d_ws, size_t ws_size,
                              hipStream_t stream) {
  const float* x     = (const float*)d_in[0];
  const float* wgt   = (const float*)d_in[1];
  const float* bias  = (const float*)d_in[2];
  const int*   terr  = (const int*)d_in[3];
  const int*   tfeat = (const int*)d_in[4];
  float*       out   = (float*)d_out;

  const int T       = in_sizes[3];
  const int i_first = out_size / (COUT * HH * WW) - T;  // host-derived, capture-safe

  const size_t packBytes = (size_t)TAPS * COUT * CIN * sizeof(float);
  const bool   packed    = (ws_size >= packBytes);

  if (packed) {
    pack_weight<<<(TAPS * COUT * CIN + 255) / 256, 256, 0, stream>>>(
        wgt, (float*)d_ws);
  }

  if (i_first > 0) {
    dim3 grid(WW / SPW, HH, i_first);
    if (packed) {
      conv_full_wmma<true><<<grid, 128, 0, stream>>>(x, (const float*)d_ws, bias, out);
    } else {
      conv_full_wmma<false><<<grid, 128, 0, stream>>>(x, wgt, bias, out);
    }
  }

  if (T > 0) {
    conv_terms_wmma<<<dim3(WW / 32, HH, T), 256, 0, stream>>>(
        x, wgt, bias, terr, tfeat, out, i_first);
  }
}